// HiddenTreeMarkovModel_37615323578473
// MI455X (gfx1250) — compile-verified
//
#include <hip/hip_runtime.h>
#include <math.h>

typedef __attribute__((ext_vector_type(2))) float v2f;
typedef __attribute__((ext_vector_type(8))) float v8f;

// Problem constants
#define T_TREES 128
#define DEPTH   11
#define CC      8
#define LL_POS  2
#define MM      256
#define GG      4
// DIM = 128*(2^12-1)
#define DIM     524160

// Workspace layout (in floats):
//  Wt  : [2][32][32]  (W transposed: Wt[p][j][k] = SP[p][g]*A[c,d,p,g], j=g*8+c, k=g*8+d, block-diag in g)
//  Bme : [256][32]    (Bme[m][g*8+c] = softmax_m(lambda_B)[c,m,g])
//  Pie : [2][32]      (Pie[p][g*8+c] = softmax_c(lambda_Pi)[c,p,g])
//  ll  : [DIM][4]
//  beta: [DIM][32]    (beta[node][g*8+c])
#define WT_OFF   0
#define BME_OFF  2048
#define PIE_OFF  10240
#define LLB_OFF  10304
#define BETA_OFF 2106944   // 10304 + DIM*4

// ---------------------------------------------------------------------------
// Kernel 1: softmaxes + build W (tiny; one workgroup)
// ---------------------------------------------------------------------------
__global__ void htmm_precompute(const float* __restrict__ lA,
                                const float* __restrict__ lB,
                                const float* __restrict__ lPi,
                                const float* __restrict__ lSP,
                                float* __restrict__ Wt,
                                float* __restrict__ Bme,
                                float* __restrict__ Pie) {
  __shared__ float sSP[8];  // [p][g]
  int tid = threadIdx.x;

  // zero W (block-diagonal fill below only writes diag blocks)
  for (int i = tid; i < 2048; i += 256) Wt[i] = 0.f;

  if (tid < 4) {  // SP: softmax over p (L=2), per g
    int g = tid;
    float a0 = lSP[0 * 4 + g], a1 = lSP[1 * 4 + g];
    float m = fmaxf(a0, a1);
    float e0 = expf(a0 - m), e1 = expf(a1 - m);
    float s = e0 + e1;
    sSP[0 * 4 + g] = e0 / s;
    sSP[1 * 4 + g] = e1 / s;
  }
  __syncthreads();

  if (tid < 64) {  // A: softmax over c, per (d,p,g); fold in SP
    int g = tid & 3, p = (tid >> 2) & 1, d = tid >> 3;
    float e[8];
    float m = -1e30f;
#pragma unroll
    for (int c = 0; c < 8; ++c) {
      e[c] = lA[((c * 8 + d) * 2 + p) * 4 + g];
      m = fmaxf(m, e[c]);
    }
    float s = 0.f;
#pragma unroll
    for (int c = 0; c < 8; ++c) { e[c] = expf(e[c] - m); s += e[c]; }
    float sp = sSP[p * 4 + g] / s;
#pragma unroll
    for (int c = 0; c < 8; ++c)
      Wt[p * 1024 + (g * 8 + c) * 32 + (g * 8 + d)] = sp * e[c];
  } else if (tid < 96) {  // B: softmax over m, per (c,g)
    int t2 = tid - 64;
    int g = t2 & 3, c = t2 >> 2;
    float m = -1e30f;
    for (int mm = 0; mm < 256; ++mm)
      m = fmaxf(m, lB[(c * 256 + mm) * 4 + g]);
    float s = 0.f;
    for (int mm = 0; mm < 256; ++mm)
      s += expf(lB[(c * 256 + mm) * 4 + g] - m);
    float inv = 1.f / s;
    for (int mm = 0; mm < 256; ++mm)
      Bme[mm * 32 + g * 8 + c] = expf(lB[(c * 256 + mm) * 4 + g] - m) * inv;
  } else if (tid < 104) {  // Pi: softmax over c, per (p,g)
    int t2 = tid - 96;
    int g = t2 & 3, p = t2 >> 2;
    float e[8];
    float m = -1e30f;
#pragma unroll
    for (int c = 0; c < 8; ++c) {
      e[c] = lPi[(c * 2 + p) * 4 + g];
      m = fmaxf(m, e[c]);
    }
    float s = 0.f;
#pragma unroll
    for (int c = 0; c < 8; ++c) { e[c] = expf(e[c] - m); s += e[c]; }
#pragma unroll
    for (int c = 0; c < 8; ++c) Pie[p * 32 + g * 8 + c] = e[c] / s;
  }
}

// ---------------------------------------------------------------------------
// Kernel 2: leaves — beta = normalize(Pi*B), ll = log(nu). One lane per leaf.
// ---------------------------------------------------------------------------
__global__ void htmm_leaves(const int* __restrict__ pos,
                            const int* __restrict__ x,
                            const float* __restrict__ Pie,
                            const float* __restrict__ Bme,
                            float* __restrict__ beta,
                            float* __restrict__ ll,
                            int leaf_off, int n_leaf) {
  int n = blockIdx.x * blockDim.x + threadIdx.x;
  if (n >= n_leaf) return;
  int node = leaf_off + n;
  const float* pr = Pie + pos[node] * 32;
  const float* br = Bme + x[node] * 32;
  float v[32];
  float nu[4] = {0.f, 0.f, 0.f, 0.f};
#pragma unroll
  for (int j = 0; j < 32; ++j) {
    v[j] = pr[j] * br[j];
    nu[j >> 3] += v[j];
  }
  float rnu[4];
#pragma unroll
  for (int g = 0; g < 4; ++g) rnu[g] = 1.f / nu[g];
  float* out = beta + (size_t)node * 32;
#pragma unroll
  for (int j4 = 0; j4 < 8; ++j4) {
    float4 o;
    o.x = v[j4 * 4 + 0] * rnu[(j4 * 4 + 0) >> 3];
    o.y = v[j4 * 4 + 1] * rnu[(j4 * 4 + 1) >> 3];
    o.z = v[j4 * 4 + 2] * rnu[(j4 * 4 + 2) >> 3];
    o.w = v[j4 * 4 + 3] * rnu[(j4 * 4 + 3) >> 3];
    *(float4*)(out + j4 * 4) = o;
  }
  float4 lo;
  lo.x = logf(nu[0]); lo.y = logf(nu[1]); lo.z = logf(nu[2]); lo.w = logf(nu[3]);
  *(float4*)(ll + (size_t)node * 4) = lo;
}

// ---------------------------------------------------------------------------
// Kernel 3: one bottom-up level. Each wave: 16 children -> 8 parents.
// GEMM via V_WMMA_F32_16X16X4_F32: rows = child nodes, cols = j=(g*8+c),
// K = k=(g*8+d) in 8 chunks of 4. Both pos-variants computed; per-row select.
// ---------------------------------------------------------------------------
__global__ void htmm_level(const int* __restrict__ pos,
                           const int* __restrict__ x,
                           const float* __restrict__ Wt,
                           const float* __restrict__ Bme,
                           float* __restrict__ beta,
                           float* __restrict__ ll,
                           int ch_off, int pa_off, int n_ch) {
  int wave = (blockIdx.x * blockDim.x + threadIdx.x) >> 5;
  int lane = threadIdx.x & 31;
  int cb = wave * 16;            // first child (within level)
  if (cb >= n_ch) return;        // wave-uniform: EXEC all-ones for WMMA
  int h = lane >> 4;             // half-wave
  int ln = lane & 15;

  // A operand: beta rows of the 16 children. Lane layout (16x4 f32 A):
  // lanes 0-15: (M=ln, K=4kk+{0,1}); lanes 16-31: (M=ln, K=4kk+{2,3}).
  v2f a[8];
  const float* brow = beta + (size_t)(ch_off + cb + ln) * 32 + 2 * h;
#pragma unroll
  for (int kk = 0; kk < 8; ++kk) a[kk] = *(const v2f*)(brow + kk * 4);

  // B operand: W slices (4x16 f32 B): lane -> (K=4kk+2h+{0,1}, N=t*16+ln).
  v2f b[2][2][8];
#pragma unroll
  for (int p = 0; p < 2; ++p)
#pragma unroll
    for (int t = 0; t < 2; ++t) {
      const float* wrow = Wt + p * 1024 + (t * 16 + ln) * 32 + 2 * h;
#pragma unroll
      for (int kk = 0; kk < 8; ++kk)
        b[p][t][kk] = *(const v2f*)(wrow + kk * 4);
    }

  v8f acc[2][2];
  v8f zero = {0.f, 0.f, 0.f, 0.f, 0.f, 0.f, 0.f, 0.f};
#pragma unroll
  for (int p = 0; p < 2; ++p)
#pragma unroll
    for (int t = 0; t < 2; ++t) acc[p][t] = zero;

#pragma unroll
  for (int kk = 0; kk < 8; ++kk) {
#pragma unroll
    for (int p = 0; p < 2; ++p)
#pragma unroll
      for (int t = 0; t < 2; ++t)
        acc[p][t] = __builtin_amdgcn_wmma_f32_16x16x4_f32(
            false, a[kk], false, b[p][t][kk], (short)0, acc[p][t], false, false);
  }

  // Per-row pos select. D layout: VGPR r holds (node = cb + r + 8h, j = t*16+ln).
  int pos_l = pos[ch_off + cb + ln];
  int pr[8];
#pragma unroll
  for (int r = 0; r < 8; ++r) pr[r] = __shfl(pos_l, r + 8 * h, 32);

  // Sibling-pair fold: parent q gets rows 2q, 2q+1.
  float pv[2][4];
#pragma unroll
  for (int t = 0; t < 2; ++t)
#pragma unroll
    for (int q = 0; q < 4; ++q) {
      float s0 = pr[2 * q]     ? acc[1][t][2 * q]     : acc[0][t][2 * q];
      float s1 = pr[2 * q + 1] ? acc[1][t][2 * q + 1] : acc[0][t][2 * q + 1];
      pv[t][q] = s0 + s1;
    }

  int pb = cb >> 1;
  int x_l = x[pa_off + pb + (lane & 7)];
#pragma unroll
  for (int q = 0; q < 4; ++q) {
    int xq = __shfl(x_l, q + 4 * h, 32);
#pragma unroll
    for (int t = 0; t < 2; ++t) {
      float v = pv[t][q] * Bme[xq * 32 + t * 16 + ln];
      float nu = v;
      nu += __shfl_xor(nu, 1, 32);
      nu += __shfl_xor(nu, 2, 32);
      nu += __shfl_xor(nu, 4, 32);   // sum over c within each g-group of 8 lanes
      int pa = pa_off + pb + q + 4 * h;
      beta[(size_t)pa * 32 + t * 16 + ln] = v / nu;
      if ((ln & 7) == 0)
        ll[(size_t)pa * 4 + 2 * t + (ln >> 3)] = logf(nu);
    }
  }
}

// ---------------------------------------------------------------------------
// Kernel 4: deterministic per-tree reduction of ll (no atomics).
// ---------------------------------------------------------------------------
__global__ void htmm_reduce(const float* __restrict__ ll, float* __restrict__ out) {
  int t = blockIdx.x;
  int tid = threadIdx.x;
  float acc[4] = {0.f, 0.f, 0.f, 0.f};
  for (int l = 0; l <= DEPTH; ++l) {
    int base = 128 * ((1 << l) - 1) + (t << l);
    int cnt = 1 << l;
    for (int i = tid; i < cnt; i += 256) {
      const float* p = ll + (size_t)(base + i) * 4;
      acc[0] += p[0]; acc[1] += p[1]; acc[2] += p[2]; acc[3] += p[3];
    }
  }
  __shared__ float s[1024];
#pragma unroll
  for (int g = 0; g < 4; ++g) s[tid * 4 + g] = acc[g];
  __syncthreads();
  for (int o = 128; o > 0; o >>= 1) {
    if (tid < o) {
#pragma unroll
      for (int g = 0; g < 4; ++g) s[tid * 4 + g] += s[(tid + o) * 4 + g];
    }
    __syncthreads();
  }
  if (tid < 4) out[t * 4 + tid] = s[tid];
}

// ---------------------------------------------------------------------------
extern "C" void kernel_launch(void* const* d_in, const int* in_sizes, int n_in,
                              void* d_out, int out_size, void* d_ws, size_t ws_size,
                              hipStream_t stream) {
  const float* lA  = (const float*)d_in[0];
  const float* lB  = (const float*)d_in[1];
  const float* lPi = (const float*)d_in[2];
  const float* lSP = (const float*)d_in[3];
  const int*   pos = (const int*)d_in[4];
  const int*   x   = (const int*)d_in[5];
  float* out = (float*)d_out;
  float* ws  = (float*)d_ws;

  float* Wt   = ws + WT_OFF;
  float* Bme  = ws + BME_OFF;
  float* Pie  = ws + PIE_OFF;
  float* llb  = ws + LLB_OFF;
  float* beta = ws + BETA_OFF;

  htmm_precompute<<<1, 256, 0, stream>>>(lA, lB, lPi, lSP, Wt, Bme, Pie);

  // leaves: level 11, base 128*(2^11-1)=262016, count 128*2^11=262144
  htmm_leaves<<<1024, 256, 0, stream>>>(pos, x, Pie, Bme, beta, llb, 262016, 262144);

  for (int l = DEPTH; l >= 1; --l) {
    int n_ch   = 128 << l;
    int ch_off = 128 * ((1 << l) - 1);
    int pa_off = 128 * ((1 << (l - 1)) - 1);
    int blocks = n_ch / 128;  // 8 waves/block, 16 children per wave
    htmm_level<<<blocks, 256, 0, stream>>>(pos, x, Wt, Bme, beta, llb,
                                           ch_off, pa_off, n_ch);
  }

  htmm_reduce<<<T_TREES, 256, 0, stream>>>(llb, out);
}